// GCNLayer_6622839571277
// MI455X (gfx1250) — compile-verified
//
#include <hip/hip_runtime.h>

typedef float v2f __attribute__((ext_vector_type(2)));
typedef float v8f __attribute__((ext_vector_type(8)));

#define N_NODES   100000
#define N_EDGES   1600000
#define IN_FEATS  256
#define OUT_FEATS 128

// -------------------------------------------------------------------------
// Kernel 1: hn[r, 0:128] = (h[r, 0:256] @ W[0:256, 0:128]) * norm[r]
// One wave per 16-row strip; 8 fp32 WMMA accumulators cover all 128 columns.
// fp32 WMMA (V_WMMA_F32_16X16X4_F32): memory-bound GEMM, keep full precision.
// -------------------------------------------------------------------------
__global__ __launch_bounds__(128)
void gcn_gemm_wmma(const float* __restrict__ h,
                   const float* __restrict__ W,
                   const float* __restrict__ norm,
                   float* __restrict__ hn) {
  const int lane  = threadIdx.x & 31;
  const int wave  = threadIdx.x >> 5;
  const int tile  = blockIdx.x * 4 + wave;         // 16-row tile index
  if (tile >= (N_NODES / 16)) return;
  const int row0  = tile * 16;
  const int m     = lane & 15;                     // row within tile / column within n-tile
  const int khalf = lane >> 4;                     // 0: lanes 0-15, 1: lanes 16-31

  v8f acc[8];
#pragma unroll
  for (int j = 0; j < 8; ++j) acc[j] = (v8f)0.0f;

  const float* hrow = h + (size_t)(row0 + m) * IN_FEATS;

  for (int k0 = 0; k0 < IN_FEATS; k0 += 4) {
    // A fragment (16x4): VGPR v, half-wave hh -> K = k0 + 2*hh + v  (contiguous pair)
    v2f a = *(const v2f*)(hrow + k0 + 2 * khalf);
#pragma unroll
    for (int j = 0; j < 8; ++j) {
      const int n = j * 16 + m;                    // output column
      v2f b;                                       // B fragment (4x16), row-striped
      b.x = W[(k0 + 2 * khalf    ) * OUT_FEATS + n];
      b.y = W[(k0 + 2 * khalf + 1) * OUT_FEATS + n];
      acc[j] = __builtin_amdgcn_wmma_f32_16x16x4_f32(
          /*neg_a=*/false, a, /*neg_b=*/false, b,
          /*c_mod=*/(short)0, acc[j], /*reuse_a=*/false, /*reuse_b=*/false);
    }
  }

  // D layout: VGPR v, lane -> row = row0 + v + 8*khalf, col = j*16 + m.
  float nrm[8];
#pragma unroll
  for (int v = 0; v < 8; ++v) nrm[v] = norm[row0 + v + 8 * khalf];

#pragma unroll
  for (int j = 0; j < 8; ++j) {
#pragma unroll
    for (int v = 0; v < 8; ++v) {
      const int r = row0 + v + 8 * khalf;
      hn[(size_t)r * OUT_FEATS + j * 16 + m] = acc[j][v] * nrm[v];
    }
  }
}

// -------------------------------------------------------------------------
// Kernel 0: zero the accumulator (d_out) — 12.8M floats as float4.
// -------------------------------------------------------------------------
__global__ __launch_bounds__(256)
void gcn_zero(float4* __restrict__ p) {
  const size_t i = (size_t)blockIdx.x * blockDim.x + threadIdx.x;
  p[i] = make_float4(0.f, 0.f, 0.f, 0.f);
}

// -------------------------------------------------------------------------
// Kernel 2: edge scatter-add. agg[dst[e], t] += hn[src[e], t].
// Both hn and agg fit in the 192MB L2 -> gather + L2 atomics stay on-chip.
// Hardware f32 atomic via inline asm (no CAS loop), device scope.
// -------------------------------------------------------------------------
__device__ __forceinline__ void atomic_add_f32_dev(float* addr, float val) {
  asm volatile("global_atomic_add_f32 %0, %1, off scope:SCOPE_DEV"
               :: "v"(addr), "v"(val) : "memory");
}

#define EDGES_PER_BLOCK 8

__global__ __launch_bounds__(128)
void gcn_scatter(const float* __restrict__ hn,
                 const int* __restrict__ src,
                 const int* __restrict__ dst,
                 float* __restrict__ agg) {
  const int t  = threadIdx.x;                      // feature index 0..127
  const int e0 = blockIdx.x * EDGES_PER_BLOCK;
#pragma unroll
  for (int i = 0; i < EDGES_PER_BLOCK; ++i) {
    const int e = e0 + i;                          // N_EDGES % 8 == 0
    const int s = src[e];
    const int d = dst[e];
    const float v = hn[(size_t)s * OUT_FEATS + t];
    atomic_add_f32_dev(&agg[(size_t)d * OUT_FEATS + t], v);
  }
}

// -------------------------------------------------------------------------
// Kernel 3: out = relu(agg * norm[row] + bias[col]), in-place on d_out.
// -------------------------------------------------------------------------
__global__ __launch_bounds__(256)
void gcn_finalize(float* __restrict__ out,
                  const float* __restrict__ norm,
                  const float* __restrict__ bias) {
  const size_t i = (size_t)blockIdx.x * blockDim.x + threadIdx.x;
  const int r = (int)(i >> 7);                     // / OUT_FEATS
  const int c = (int)(i & (OUT_FEATS - 1));
  const float v = out[i] * norm[r] + bias[c];
  out[i] = v > 0.f ? v : 0.f;
}

// -------------------------------------------------------------------------
// Inputs (setup_inputs order): h, weight, bias, norm, src, dst.
// -------------------------------------------------------------------------
extern "C" void kernel_launch(void* const* d_in, const int* in_sizes, int n_in,
                              void* d_out, int out_size, void* d_ws, size_t ws_size,
                              hipStream_t stream) {
  const float* h    = (const float*)d_in[0];
  const float* W    = (const float*)d_in[1];
  const float* bias = (const float*)d_in[2];
  const float* norm = (const float*)d_in[3];
  const int*   src  = (const int*)d_in[4];
  const int*   dst  = (const int*)d_in[5];
  float*       out  = (float*)d_out;               // N_NODES * OUT_FEATS
  float*       hn   = (float*)d_ws;                // N_NODES * OUT_FEATS (51.2 MB)

  const int total     = N_NODES * OUT_FEATS;       // 12,800,000
  const int zeroBlks  = total / 4 / 256;           // 12,500
  const int gemmBlks  = (N_NODES / 16 + 3) / 4;    // 1,563 (4 waves/block)
  const int scatBlks  = N_EDGES / EDGES_PER_BLOCK; // 200,000
  const int finBlks   = total / 256;               // 50,000

  gcn_zero    <<<zeroBlks, 256, 0, stream>>>((float4*)out);
  gcn_gemm_wmma<<<gemmBlks, 128, 0, stream>>>(h, W, norm, hn);
  gcn_scatter <<<scatBlks, 128, 0, stream>>>(hn, src, dst, out);
  gcn_finalize<<<finBlks, 256, 0, stream>>>(out, norm, bias);
}